// QGT_6Qubit_Model_47777216201065
// MI455X (gfx1250) — compile-verified
//
#include <hip/hip_runtime.h>
#include <hip/hip_bf16.h>
#include <math.h>

// ---------------------------------------------------------------------------
// QGT 6-qubit quantum graph transformer for MI455X (gfx1250, wave32).
// QFT64 (the only real GEMM) runs on the WMMA pipe as f32 16x16x4 tiles out
// of LDS, with dual accumulators per tile to break WMMA dependency chains
// and 4 waves/block for latency hiding.  Edge softmax/scatter uses L2
// atomics (working set < 192MB L2).  All reductions are wave32 shfl trees.
// ---------------------------------------------------------------------------

#define NN   100000
#define NE   1200000
#define NG   512
#define EMB  64
#define NLAY 2
#define QKD  18
#define LDSW 65           // padded row stride (floats) to dodge bank conflicts

typedef float v2f __attribute__((ext_vector_type(2)));
typedef float v8f __attribute__((ext_vector_type(8)));

// ---- WMMA f32 16x16x4 fragment helpers (layouts per CDNA5 ISA) --
// A (16x4 f32): lanes 0-15 -> M=lane, VGPR0=K0,VGPR1=K1; lanes 16-31 -> K2,K3
__device__ __forceinline__ v2f ldA(const float* S, int m0, int k0, int lane) {
  int r  = m0 + (lane & 15);
  int kk = k0 + ((lane >> 4) << 1);
  v2f a; a.x = S[r * LDSW + kk]; a.y = S[r * LDSW + kk + 1]; return a;
}
// B (4x16 f32): mirrored — lanes 0-15 row k0/k0+1, lanes 16-31 row k0+2/k0+3
__device__ __forceinline__ v2f ldB(const float* F, int k0, int n0, int lane) {
  int n  = n0 + (lane & 15);
  int kk = k0 + ((lane >> 4) << 1);
  v2f b; b.x = F[kk * LDSW + n]; b.y = F[(kk + 1) * LDSW + n]; return b;
}
// D (16x16 f32): VGPR r: lanes 0-15 -> M=r,N=lane ; lanes 16-31 -> M=8+r
__device__ __forceinline__ void stD(float* O, v8f d, int m0, int n0, int lane) {
  int n  = n0 + (lane & 15);
  int mb = m0 + ((lane >> 4) << 3);
#pragma unroll
  for (int r = 0; r < 8; ++r) O[(mb + r) * LDSW + n] = d[r];
}

// ---------------------------------------------------------------------------
// One 6-qubit state sim for 64 nodes (block of 128 threads / 4 waves):
// product state + CNOT ring + RY per-thread in LDS (waves 0-1), then QFT64
// via WMMA (all 4 waves, 4 tile-jobs each, dual accumulators), then X/Y/Z
// expvals (wave 0).
// ---------------------------------------------------------------------------
__device__ void run_state(float* Sr, float* Si, const float* Fr, const float* Fi,
                          float* Or, float* Oi, const float* ang, const float* w,
                          int t, int lane, int wave, int blockbase, float temp,
                          const float* pad, float* out)
{
  __syncthreads();                 // staging buffer (Sr) free to overwrite

  if (t < 64) {
    float* Srp = Sr + t * LDSW;
    float* Sip = Si + t * LDSW;

    // ---- product state: G = Rz(w2) Ry(w1) Rx(w0) on (cos a/2, sin a/2)
    Srp[0] = 1.f; Sip[0] = 0.f;
    int size = 1;
    for (int q = 0; q < 6; ++q) {
      float ha = 0.5f * ang[q];
      float u0 = cosf(ha), u1 = sinf(ha);
      float cx = cosf(0.5f * w[q * 4 + 0]), sx = sinf(0.5f * w[q * 4 + 0]);
      float cy = cosf(0.5f * w[q * 4 + 1]), sy = sinf(0.5f * w[q * 4 + 1]);
      float cz = cosf(0.5f * w[q * 4 + 2]), sz = sinf(0.5f * w[q * 4 + 2]);
      float p0r = cx * u0, p0i = -sx * u1;          // Rx
      float p1r = cx * u1, p1i = -sx * u0;
      float q0r = cy * p0r - sy * p1r, q0i = cy * p0i - sy * p1i;   // Ry
      float q1r = sy * p0r + cy * p1r, q1i = sy * p0i + cy * p1i;
      float v0r = q0r * cz + q0i * sz, v0i = q0i * cz - q0r * sz;   // Rz e^{-i}
      float v1r = q1r * cz - q1i * sz, v1i = q1i * cz + q1r * sz;   // e^{+i}
      for (int j = size - 1; j >= 0; --j) {         // append qubit as LSB
        float pr = Srp[j], pi = Sip[j];
        Srp[2 * j + 1] = pr * v1r - pi * v1i; Sip[2 * j + 1] = pr * v1i + pi * v1r;
        Srp[2 * j]     = pr * v0r - pi * v0i; Sip[2 * j]     = pr * v0i + pi * v0r;
      }
      size <<= 1;
    }
    // ---- CNOT ring (qubit q lives at bit 5-q)
    for (int c = 0; c < 6; ++c) {
      int tq = (c + 1) % 6;
      int cm = 1 << (5 - c), tm = 1 << (5 - tq);
      for (int i = 0; i < 64; ++i)
        if ((i & cm) && !(i & tm)) {
          int j = i | tm;
          float ar = Srp[i], ai = Sip[i];
          Srp[i] = Srp[j]; Sip[i] = Sip[j];
          Srp[j] = ar;     Sip[j] = ai;
        }
    }
    // ---- final RY layer
    for (int q = 0; q < 6; ++q) {
      float c3 = cosf(0.5f * w[q * 4 + 3]), s3 = sinf(0.5f * w[q * 4 + 3]);
      int qm = 1 << (5 - q);
      for (int i = 0; i < 64; ++i)
        if (!(i & qm)) {
          int j = i | qm;
          float a0r = Srp[i], a0i = Sip[i], a1r = Srp[j], a1i = Sip[j];
          Srp[i] = c3 * a0r - s3 * a1r; Sip[i] = c3 * a0i - s3 * a1i;
          Srp[j] = s3 * a0r + c3 * a1r; Sip[j] = s3 * a0i + c3 * a1i;
        }
    }
  }
  __syncthreads();

  // ---- QFT64 via WMMA: OUT = S @ F  (F symmetric), complex -> 4 real GEMMs.
  // Staged in two M-halves so the O buffers are only 32 rows (LDS budget).
  for (int mb = 0; mb < 64; mb += 32) {
    for (int jj = 0; jj < 4; ++jj) {
      int job  = wave * 4 + jj;             // 16 jobs over 4 waves
      int comp = job & 1;                   // 0: real out, 1: imag out
      int tl   = job >> 1;
      int m0   = mb + (tl & 1) * 16;
      int n0   = (tl >> 1) * 16;
      v8f acc0 = {};                        // two independent WMMA chains
      v8f acc1 = {};
      if (comp == 0) {                      // Or = Sr*Fr - Si*Fi
#pragma unroll
        for (int k0 = 0; k0 < 64; k0 += 4) {
          v2f a = ldA(Sr, m0, k0, lane);
          v2f b = ldB(Fr, k0, n0, lane);
          acc0 = __builtin_amdgcn_wmma_f32_16x16x4_f32(false, a, false, b,
                                                       (short)0, acc0, false, false);
        }
#pragma unroll
        for (int k0 = 0; k0 < 64; k0 += 4) {
          v2f a = ldA(Si, m0, k0, lane);
          v2f b = ldB(Fi, k0, n0, lane);
          b = -b;                           // f32 WMMA has no A/B neg bits
          acc1 = __builtin_amdgcn_wmma_f32_16x16x4_f32(false, a, false, b,
                                                       (short)0, acc1, false, false);
        }
        stD(Or, acc0 + acc1, m0 - mb, n0, lane);
      } else {                              // Oi = Sr*Fi + Si*Fr
#pragma unroll
        for (int k0 = 0; k0 < 64; k0 += 4) {
          v2f a = ldA(Sr, m0, k0, lane);
          v2f b = ldB(Fi, k0, n0, lane);
          acc0 = __builtin_amdgcn_wmma_f32_16x16x4_f32(false, a, false, b,
                                                       (short)0, acc0, false, false);
        }
#pragma unroll
        for (int k0 = 0; k0 < 64; k0 += 4) {
          v2f a = ldA(Si, m0, k0, lane);
          v2f b = ldB(Fr, k0, n0, lane);
          acc1 = __builtin_amdgcn_wmma_f32_16x16x4_f32(false, a, false, b,
                                                       (short)0, acc1, false, false);
        }
        stD(Oi, acc0 + acc1, m0 - mb, n0, lane);
      }
    }
    __syncthreads();
    // ---- measurement for the 32 rows just produced (wave 0)
    if (t < 32) {
      int node = blockbase + mb + t;
      if (node < NN) {
        const float* orow = Or + t * LDSW;
        const float* irow = Oi + t * LDSW;
        float sc = pad[node] * temp;
        for (int q = 0; q < 6; ++q) {
          int qm = 1 << (5 - q);
          float re = 0.f, im = 0.f, zz = 0.f;
          for (int i = 0; i < 64; ++i)
            if (!(i & qm)) {
              int j = i | qm;
              float a0r = orow[i], a0i = irow[i], a1r = orow[j], a1i = irow[j];
              re += a0r * a1r + a0i * a1i;
              im += a0r * a1i - a0i * a1r;
              zz += a0r * a0r + a0i * a0i - a1r * a1r - a1i * a1i;
            }
          out[node * QKD + q]      = 2.f * re * sc;
          out[node * QKD + 6 + q]  = 2.f * im * sc;
          out[node * QKD + 12 + q] = zz * sc;
        }
      }
    }
    __syncthreads();
  }
}

// ---------------------------------------------------------------------------
// Per-node kernel: coalesced x staging -> angles -> two quantum sims -> Q,K.
// 64 nodes/block, 128 threads (4 waves); WMMA runs with full EXEC.
// ---------------------------------------------------------------------------
__global__ void __launch_bounds__(128) pqc_kernel(
    const float* __restrict__ xin, const float* __restrict__ pad,
    const float* __restrict__ qaeW, const float* __restrict__ qaeb,
    const float* __restrict__ pqW, const float* __restrict__ pqb,
    const float* __restrict__ wqv,
    const float* __restrict__ pkW, const float* __restrict__ pkb,
    const float* __restrict__ wkv,
    const float* __restrict__ lsc, int layer,
    float* __restrict__ Qout, float* __restrict__ Kout)
{
  extern __shared__ float smem[];
  float* Sr = smem;                  // 64 x LDSW (also x staging tile)
  float* Si = Sr + 64 * LDSW;
  float* Fr = Si + 64 * LDSW;
  float* Fi = Fr + 64 * LDSW;
  float* Or = Fi + 64 * LDSW;        // 32 x LDSW
  float* Oi = Or + 32 * LDSW;

  const int t         = threadIdx.x;          // 0..127
  const int lane      = t & 31;
  const int wave      = t >> 5;
  const int blockbase = blockIdx.x * 64;

  // ---- Fourier matrix B[k][j] = exp(2pi i kj/64)/8 (symmetric):
  // 128 threads fill 64 rows, half a row each.
  {
    int r  = t & 63;
    int j0 = (t >> 6) * 32;
    for (int j = j0; j < j0 + 32; ++j) {
      int ph = (r * j) & 63;
      float s, c;
      sincosf((float)ph * 0.0981747704247f, &s, &c);   // 2*pi/64
      Fr[r * LDSW + j] = c * 0.125f;
      Fi[r * LDSW + j] = s * 0.125f;
    }
  }

  // ---- coalesced staging of this block's 64x64 x tile into Sr,
  // stored transposed (Sr[dim*LDSW + node_local]) so both the coalesced
  // global load and the per-dim broadcast read are bank-conflict-free.
  {
    long long base = (long long)blockbase * EMB;
    for (int k2 = 0; k2 < 32; ++k2) {
      int idx = k2 * 128 + t;                  // 4096 elements / block
      long long g = base + idx;
      if (g > (long long)NN * EMB - 1) g = (long long)NN * EMB - 1;
      int nl = idx >> 6, d = idx & 63;
      Sr[d * LDSW + nl] = xin[g];
    }
  }
  __syncthreads();

  // ---- qae latent + projections -> angles (waves 0-1, one node each)
  float aq[6] = {0, 0, 0, 0, 0, 0};
  float ak[6] = {0, 0, 0, 0, 0, 0};
  if (t < 64) {
    float lat[QKD];
#pragma unroll
    for (int j = 0; j < QKD; ++j) lat[j] = qaeb[j];
    for (int d = 0; d < EMB; ++d) {
      float xv = Sr[d * LDSW + t];
#pragma unroll
      for (int j = 0; j < QKD; ++j) lat[j] += xv * qaeW[d * QKD + j];
    }
#pragma unroll
    for (int j = 0; j < QKD; ++j) lat[j] = tanhf(lat[j]);

    const float PI = 3.14159265358979f;
    const float* pqWl = pqW + layer * QKD * 6;
    const float* pkWl = pkW + layer * QKD * 6;
#pragma unroll
    for (int k = 0; k < 6; ++k) {
      float a = pqb[layer * 6 + k], b = pkb[layer * 6 + k];
      for (int j = 0; j < QKD; ++j) {
        a += lat[j] * pqWl[j * 6 + k];
        b += lat[j] * pkWl[j * 6 + k];
      }
      aq[k] = tanhf(a) * PI;
      ak[k] = tanhf(b) * PI;
    }
  }
  float temp = expf(lsc[layer]);

  run_state(Sr, Si, Fr, Fi, Or, Oi, aq, wqv + layer * 24,
            t, lane, wave, blockbase, temp, pad, Qout);
  run_state(Sr, Si, Fr, Fi, Or, Oi, ak, wkv + layer * 24,
            t, lane, wave, blockbase, temp, pad, Kout);
}

// ---------------------------------------------------------------------------
// Edge softmax + scatter pipeline
// ---------------------------------------------------------------------------
__device__ __forceinline__ unsigned fkey(float f) {
  unsigned u = __float_as_uint(f);
  return (u & 0x80000000u) ? ~u : (u | 0x80000000u);  // order-preserving
}
__device__ __forceinline__ float funkey(unsigned k) {
  return (k & 0x80000000u) ? __uint_as_float(k & 0x7FFFFFFFu) : __uint_as_float(~k);
}

__global__ void init_layer_kernel(float* m, unsigned* nmax, float* nden) {
  int i = blockIdx.x * blockDim.x + threadIdx.x;
  if (i < NN * EMB) m[i] = 0.f;
  if (i < NN) { nmax[i] = 0u; nden[i] = 0.f; }
}

__global__ void logits_kernel(const long long* __restrict__ ei,
                              const float* __restrict__ Q, const float* __restrict__ K,
                              float* __restrict__ lg, unsigned* __restrict__ nmax) {
  int e = blockIdx.x * blockDim.x + threadIdx.x;
  if (e >= NE) return;
  int s = (int)ei[e], d = (int)ei[NE + e];
  float acc = 0.f;
#pragma unroll
  for (int j = 0; j < QKD; ++j) acc += Q[d * QKD + j] * K[s * QKD + j];
  float v = acc * 0.235702260396f;          // 1/sqrt(18)
  lg[e] = v;
  atomicMax(&nmax[d], fkey(v));
}

__global__ void exp_kernel(const long long* __restrict__ ei, float* __restrict__ lg,
                           const unsigned* __restrict__ nmax, float* __restrict__ nden) {
  int e = blockIdx.x * blockDim.x + threadIdx.x;
  if (e >= NE) return;
  int d = (int)ei[NE + e];
  float ex = expf(lg[e] - funkey(nmax[d]));
  lg[e] = ex;
  atomicAdd(&nden[d], ex);
}

__global__ void scatter_kernel(const long long* __restrict__ ei,
                               const float* __restrict__ ex, const float* __restrict__ nden,
                               const float* __restrict__ xcur, const float* __restrict__ pad,
                               float* __restrict__ m, float* __restrict__ alpha_out) {
  int i = blockIdx.x * blockDim.x + threadIdx.x;   // NE*64 = 76.8M < 2^31
  if (i >= NE * EMB) return;
  int e = i >> 6, d = i & 63;
  int s = (int)ei[e], dn = (int)ei[NE + e];
  float alpha = ex[e] / (nden[dn] + 1e-16f);
  if (d == 0) alpha_out[e] = alpha;
  atomicAdd(&m[dn * EMB + d], xcur[s * EMB + d] * pad[s] * alpha);
}

// one wave per node: residual + mask + layernorm + relu (wave32 shfl tree)
__global__ void update_kernel(float* __restrict__ xcur, const float* __restrict__ m,
                              const float* __restrict__ pad, const float* __restrict__ mixp,
                              const float* __restrict__ lng, const float* __restrict__ lnb,
                              int layer) {
  int wid  = (blockIdx.x * blockDim.x + threadIdx.x) >> 5;
  int lane = threadIdx.x & 31;
  if (wid >= NN) return;
  float mk = pad[wid], mx = mixp[layer];
  float v0 = (xcur[wid * 64 + lane]      + mx * m[wid * 64 + lane])      * mk;
  float v1 = (xcur[wid * 64 + 32 + lane] + mx * m[wid * 64 + 32 + lane]) * mk;
  float s = v0 + v1;
  for (int o = 16; o; o >>= 1) s += __shfl_xor(s, o, 32);
  float mean = s * (1.f / 64.f);
  float d0 = v0 - mean, d1 = v1 - mean;
  float vs = d0 * d0 + d1 * d1;
  for (int o = 16; o; o >>= 1) vs += __shfl_xor(vs, o, 32);
  float rstd = rsqrtf(vs * (1.f / 64.f) + 1e-5f);
  float o0 = fmaxf(d0 * rstd * lng[layer * 64 + lane]      + lnb[layer * 64 + lane], 0.f);
  float o1 = fmaxf(d1 * rstd * lng[layer * 64 + 32 + lane] + lnb[layer * 64 + 32 + lane], 0.f);
  xcur[wid * 64 + lane]      = o0;
  xcur[wid * 64 + 32 + lane] = o1;
}

__global__ void init_pool_kernel(float* gsum, float* gcnt) {
  int i = blockIdx.x * blockDim.x + threadIdx.x;
  if (i < NG * EMB) gsum[i] = 0.f;
  if (i < NG) gcnt[i] = 0.f;
}

__global__ void pool_kernel(const float* __restrict__ xcur, const float* __restrict__ pad,
                            const long long* __restrict__ batch,
                            float* __restrict__ gsum, float* __restrict__ gcnt) {
  int wid  = (blockIdx.x * blockDim.x + threadIdx.x) >> 5;
  int lane = threadIdx.x & 31;
  if (wid >= NN) return;
  int b = (int)batch[wid];
  float mk = pad[wid];
  atomicAdd(&gsum[b * 64 + lane],      xcur[wid * 64 + lane] * mk);
  atomicAdd(&gsum[b * 64 + 32 + lane], xcur[wid * 64 + 32 + lane] * mk);
  if (lane == 0) atomicAdd(&gcnt[b], mk);
}

__global__ void final_kernel(const float* __restrict__ gsum, const float* __restrict__ gcnt,
                             const float* __restrict__ fcW, const float* __restrict__ fcb,
                             float* __restrict__ out) {
  int i = blockIdx.x * blockDim.x + threadIdx.x;
  if (i >= NG * 2) return;
  int g = i >> 1, c = i & 1;
  float cnt = fmaxf(gcnt[g], 1e-8f);
  float acc = fcb[c];
  for (int d = 0; d < 64; ++d) acc += (gsum[g * 64 + d] / cnt) * fcW[d * 2 + c];
  out[i] = acc;
}

__global__ void dst_kernel(const long long* __restrict__ ei, float* __restrict__ out) {
  int e = blockIdx.x * blockDim.x + threadIdx.x;
  if (e >= NE) return;
  out[e] = (float)ei[NE + e];
}

// ---------------------------------------------------------------------------
extern "C" void kernel_launch(void* const* d_in, const int* in_sizes, int n_in,
                              void* d_out, int out_size, void* d_ws, size_t ws_size,
                              hipStream_t stream) {
  const float*     x     = (const float*)d_in[0];
  const long long* ei    = (const long long*)d_in[1];
  const float*     pad   = (const float*)d_in[2];
  const long long* batch = (const long long*)d_in[3];
  const float* qaeW = (const float*)d_in[4];
  const float* qaeb = (const float*)d_in[5];
  const float* pqW  = (const float*)d_in[6];
  const float* pqb  = (const float*)d_in[7];
  const float* wq   = (const float*)d_in[8];
  const float* pkW  = (const float*)d_in[9];
  const float* pkb  = (const float*)d_in[10];
  const float* wk   = (const float*)d_in[11];
  const float* lsc  = (const float*)d_in[12];
  const float* mix  = (const float*)d_in[13];
  const float* lng  = (const float*)d_in[14];
  const float* lnb  = (const float*)d_in[15];
  const float* fcW  = (const float*)d_in[16];
  const float* fcb  = (const float*)d_in[17];
  float* out = (float*)d_out;

  // workspace carve-up (~71 MB)
  float* ws   = (float*)d_ws;
  float* xcur = ws;
  float* mbuf = xcur + (size_t)NN * EMB;
  float* Qb   = mbuf + (size_t)NN * EMB;
  float* Kb   = Qb + (size_t)NN * QKD;
  float* exb  = Kb + (size_t)NN * QKD;
  unsigned* nmax = (unsigned*)(exb + NE);
  float* nden = (float*)(nmax + NN);
  float* gsum = nden + NN;
  float* gcnt = gsum + NG * EMB;

  hipMemcpyAsync(xcur, x, sizeof(float) * (size_t)NN * EMB,
                 hipMemcpyDeviceToDevice, stream);

  const int SMEM = (4 * 64 * LDSW + 2 * 32 * LDSW) * (int)sizeof(float); // 83.2KB

  for (int l = 0; l < NLAY; ++l) {
    init_layer_kernel<<<(NN * EMB + 255) / 256, 256, 0, stream>>>(mbuf, nmax, nden);
    pqc_kernel<<<(NN + 63) / 64, 128, SMEM, stream>>>(
        xcur, pad, qaeW, qaeb, pqW, pqb, wq, pkW, pkb, wk, lsc, l, Qb, Kb);
    logits_kernel<<<(NE + 255) / 256, 256, 0, stream>>>(ei, Qb, Kb, exb, nmax);
    exp_kernel<<<(NE + 255) / 256, 256, 0, stream>>>(ei, exb, nmax, nden);
    scatter_kernel<<<(NE * EMB + 255) / 256, 256, 0, stream>>>(
        ei, exb, nden, xcur, pad, mbuf, out + NG * 2);
    update_kernel<<<(NN * 32 + 255) / 256, 256, 0, stream>>>(
        xcur, mbuf, pad, mix, lng, lnb, l);
  }
  init_pool_kernel<<<(NG * EMB + 255) / 256, 256, 0, stream>>>(gsum, gcnt);
  pool_kernel<<<(NN * 32 + 255) / 256, 256, 0, stream>>>(xcur, pad, batch, gsum, gcnt);
  final_kernel<<<(NG * 2 + 255) / 256, 256, 0, stream>>>(gsum, gcnt, fcW, fcb, out);
  dst_kernel<<<(NE + 255) / 256, 256, 0, stream>>>(ei, out + NG * 2 + NE);
}